// Decoder_57062935495094
// MI455X (gfx1250) — compile-verified
//
#include <hip/hip_runtime.h>
#include <math.h>

typedef __attribute__((ext_vector_type(2))) float v2f;
typedef __attribute__((ext_vector_type(8))) float v8f;
typedef __attribute__((ext_vector_type(2))) int   v2i;

#define BN   256
#define IN   256
#define HN   512
#define AN   400
#define VN   50000
#define OOVN 50
#define VP   (VN + OOVN)   // 50050
#define G4H  2048          // 4*H

#if defined(__gfx1250__) &&                                        \
    __has_builtin(__builtin_amdgcn_global_load_async_to_lds_b64) && \
    __has_builtin(__builtin_amdgcn_s_wait_asynccnt)
#define USE_ASYNC 1
#else
#define USE_ASYNC 0
#endif

#if USE_ASYNC
typedef __attribute__((address_space(1))) v2i gas_v2i;   // global <2 x i32>
typedef __attribute__((address_space(3))) v2i las_v2i;   // LDS    <2 x i32>
#endif

__device__ __forceinline__ v8f wmma4(v2f a, v2f b, v8f c) {
  // D = A(16x4 f32) * B(4x16 f32) + C(16x16 f32)
  return __builtin_amdgcn_wmma_f32_16x16x4_f32(false, a, false, b, (short)0, c,
                                               false, false);
}

__device__ __forceinline__ float sigmoidf_(float x) {
  return 1.0f / (1.0f + expf(-x));
}

// ---------------------------------------------------------------------------
// Kernel 1: gates = x0 @ W_ih^T + b_ih + b_hh      (M=256, N=2048, K=256)
// ---------------------------------------------------------------------------
__global__ __launch_bounds__(256) void gates_gemm(const float* __restrict__ x0,
                                                  const float* __restrict__ Wih,
                                                  const float* __restrict__ bih,
                                                  const float* __restrict__ bhh,
                                                  float* __restrict__ gates) {
  const int wave = threadIdx.x >> 5;
  const int lane = threadIdx.x & 31;
  const int tile = blockIdx.x * 8 + wave;       // 2048 tiles
  const int mt = tile >> 7;                     // /128  -> 0..15
  const int nt = tile & 127;                    // 0..127
  const int m0 = mt * 16, n0 = nt * 16;
  const int l16 = lane & 15, kh = lane >> 4;

  const float* arow = x0  + (m0 + l16) * IN;
  const float* brow = Wih + (n0 + l16) * IN;
  v8f acc = {};
#pragma unroll 8
  for (int k = 0; k < IN; k += 4) {
    v2f a = *(const v2f*)(arow + k + 2 * kh);
    v2f b = *(const v2f*)(brow + k + 2 * kh);
    acc = wmma4(a, b, acc);
  }
  const float bias = bih[n0 + l16] + bhh[n0 + l16];
#pragma unroll
  for (int r = 0; r < 8; ++r)
    gates[(m0 + r + 8 * kh) * G4H + n0 + l16] = acc[r] + bias;
}

// ---------------------------------------------------------------------------
// Kernel 2: LSTM activations -> h   (c = sig(i)*tanh(g); h = sig(o)*tanh(c))
// ---------------------------------------------------------------------------
__global__ __launch_bounds__(256) void lstm_act(const float* __restrict__ gates,
                                                float* __restrict__ h) {
  const int idx = blockIdx.x * 256 + threadIdx.x;   // 256*512 total
  const int b = idx >> 9;       // /512
  const int j = idx & 511;
  const float ig = gates[b * G4H + j];
  const float gg = gates[b * G4H + 2 * HN + j];
  const float og = gates[b * G4H + 3 * HN + j];
  const float c = sigmoidf_(ig) * tanhf(gg);
  h[b * HN + j] = sigmoidf_(og) * tanhf(c);
}

// ---------------------------------------------------------------------------
// Kernel 3: e = tanh(enc_state @ Wh^T + Wh_b + h @ Ws^T + Ws_b)
//           (M=256, N=400, K=512 twice). 400 tiles, 50 blocks.
// ---------------------------------------------------------------------------
__global__ __launch_bounds__(256) void attn_e(const float* __restrict__ estate,
                                              const float* __restrict__ Whw,
                                              const float* __restrict__ Whb,
                                              const float* __restrict__ h,
                                              const float* __restrict__ Wsw,
                                              const float* __restrict__ Wsb,
                                              float* __restrict__ e) {
  const int wave = threadIdx.x >> 5;
  const int lane = threadIdx.x & 31;
  const int tile = blockIdx.x * 8 + wave;       // 400 tiles
  const int mt = tile / 25;
  const int nt = tile % 25;
  const int m0 = mt * 16, n0 = nt * 16;
  const int l16 = lane & 15, kh = lane >> 4;

  v8f acc = {};
  {
    const float* arow = estate + (m0 + l16) * HN;
    const float* brow = Whw + (n0 + l16) * HN;
#pragma unroll 8
    for (int k = 0; k < HN; k += 4) {
      v2f a = *(const v2f*)(arow + k + 2 * kh);
      v2f b = *(const v2f*)(brow + k + 2 * kh);
      acc = wmma4(a, b, acc);
    }
  }
  {
    const float* arow = h + (m0 + l16) * HN;
    const float* brow = Wsw + (n0 + l16) * HN;
#pragma unroll 8
    for (int k = 0; k < HN; k += 4) {
      v2f a = *(const v2f*)(arow + k + 2 * kh);
      v2f b = *(const v2f*)(brow + k + 2 * kh);
      acc = wmma4(a, b, acc);
    }
  }
  const float bias = Whb[n0 + l16] + Wsb[n0 + l16];
#pragma unroll
  for (int r = 0; r < 8; ++r)
    e[(m0 + r + 8 * kh) * AN + n0 + l16] = tanhf(acc[r] + bias);
}

// ---------------------------------------------------------------------------
// Kernel 4: per batch row: softmax(e) * v -> attention; context = att @ EO;
//           gen = sigmoid(ctx.pg2 + x0.pg1 + h.pg3).  One block per row.
// ---------------------------------------------------------------------------
__global__ __launch_bounds__(256) void attn_ctx(const float* __restrict__ e,
                                                const float* __restrict__ vvec,
                                                const float* __restrict__ eo,
                                                const float* __restrict__ x0,
                                                const float* __restrict__ h,
                                                const float* __restrict__ pg1,
                                                const float* __restrict__ pg2,
                                                const float* __restrict__ pg3,
                                                float* __restrict__ att,
                                                float* __restrict__ ctx,
                                                float* __restrict__ gen) {
  __shared__ float s_att[AN];
  __shared__ float red[256];
  const int b = blockIdx.x;
  const int t = threadIdx.x;
  const float* erow = e + b * AN;

  // row max
  float m = -3.4e38f;
  for (int a = t; a < AN; a += 256) m = fmaxf(m, erow[a]);
  __syncthreads();
  red[t] = m; __syncthreads();
  for (int s = 128; s > 0; s >>= 1) {
    if (t < s) red[t] = fmaxf(red[t], red[t + s]);
    __syncthreads();
  }
  m = red[0];
  __syncthreads();

  // exp + sum
  float sum = 0.f;
  for (int a = t; a < AN; a += 256) {
    float ex = expf(erow[a] - m);
    s_att[a] = ex;
    sum += ex;
  }
  red[t] = sum; __syncthreads();
  for (int s = 128; s > 0; s >>= 1) {
    if (t < s) red[t] += red[t + s];
    __syncthreads();
  }
  const float inv = 1.0f / red[0];
  __syncthreads();

  for (int a = t; a < AN; a += 256) {
    float w = vvec[a] * s_att[a] * inv;
    s_att[a] = w;
    att[b * AN + a] = w;
  }
  __syncthreads();

  // context: coalesced stream over encoder_outputs[b]
  const float* eob = eo + (size_t)b * AN * AN;
  float c0 = 0.f, c1 = 0.f;
  for (int a = 0; a < AN; ++a) {
    const float w = s_att[a];
    c0 += w * eob[a * AN + t];
    if (t < AN - 256) c1 += w * eob[a * AN + 256 + t];
  }
  ctx[b * AN + t] = c0;
  if (t < AN - 256) ctx[b * AN + 256 + t] = c1;

  // gen = sigmoid(ctx.pg2 + x0.pg1 + h.pg3)
  float g = c0 * pg2[t];
  if (t < AN - 256) g += c1 * pg2[256 + t];
  g += x0[b * IN + t] * pg1[t];                 // I == 256 == blockDim
  for (int j = t; j < HN; j += 256) g += h[b * HN + j] * pg3[j];
  __syncthreads();
  red[t] = g; __syncthreads();
  for (int s = 128; s > 0; s >>= 1) {
    if (t < s) red[t] += red[t + s];
    __syncthreads();
  }
  if (t == 0) gen[b] = sigmoidf_(red[0]);
}

// ---------------------------------------------------------------------------
// Kernel 5: hidden = [ctx|h] @ fc1_w^T + fc1_b  (M=256, N=1024, K=400+512)
// ---------------------------------------------------------------------------
__global__ __launch_bounds__(256) void fc1_gemm(const float* __restrict__ ctx,
                                                const float* __restrict__ h,
                                                const float* __restrict__ W,
                                                const float* __restrict__ bias,
                                                float* __restrict__ hid) {
  const int wave = threadIdx.x >> 5;
  const int lane = threadIdx.x & 31;
  const int tile = blockIdx.x * 8 + wave;   // 1024 tiles
  const int mt = tile >> 6;                 // 0..15
  const int nt = tile & 63;                 // 0..63
  const int m0 = mt * 16, n0 = nt * 16;
  const int l16 = lane & 15, kh = lane >> 4;
  const int ldw = AN + HN;                  // 912

  v8f acc = {};
  {
    const float* arow = ctx + (m0 + l16) * AN;
    const float* brow = W + (n0 + l16) * ldw;
#pragma unroll 4
    for (int k = 0; k < AN; k += 4) {
      v2f a = *(const v2f*)(arow + k + 2 * kh);
      v2f b = *(const v2f*)(brow + k + 2 * kh);
      acc = wmma4(a, b, acc);
    }
  }
  {
    const float* arow = h + (m0 + l16) * HN;
    const float* brow = W + (n0 + l16) * ldw + AN;
#pragma unroll 4
    for (int k = 0; k < HN; k += 4) {
      v2f a = *(const v2f*)(arow + k + 2 * kh);
      v2f b = *(const v2f*)(brow + k + 2 * kh);
      acc = wmma4(a, b, acc);
    }
  }
  const float bb = bias[n0 + l16];
#pragma unroll
  for (int r = 0; r < 8; ++r)
    hid[(m0 + r + 8 * kh) * 1024 + n0 + l16] = acc[r] + bb;
}

// ---------------------------------------------------------------------------
// Kernel 6: logits = hidden @ fc2_w^T + fc2_b  -> d_out rows (stride 50050).
//
// One N-tile (16 fc2_w rows, K=1024) per block: each weight byte leaves HBM
// exactly once and is reused across all 16 M-tiles.  The weight chunk
// (16 rows x 64 k) is staged into LDS in *fragment-linear* order via CDNA5
// async global->LDS loads (double-buffered, s_wait_asynccnt + barrier), so
// each per-K-step B-fragment read is one conflict-free ds_load_b64
// (lane l -> banks 2l, 2l+1).  Wave w computes M-tiles 2w and 2w+1 sharing
// one B fragment.
// ---------------------------------------------------------------------------
#define KC 64                       // k-chunk (floats)
#define CHUNK_FLTS (16 * KC)        // 1024 floats per buffer

__device__ __forceinline__ void stage_b64(const float* __restrict__ g,
                                          float* l) {
#if USE_ASYNC
  __builtin_amdgcn_global_load_async_to_lds_b64((gas_v2i*)(g), (las_v2i*)(l),
                                                0, 0);
#else
  *(v2f*)l = *(const v2f*)g;
#endif
}

__global__ __launch_bounds__(256) void fc2_gemm(const float* __restrict__ hid,
                                                const float* __restrict__ W,
                                                const float* __restrict__ bias,
                                                float* __restrict__ out) {
  __shared__ float sb[2][CHUNK_FLTS];

  const int t = threadIdx.x;
  const int wave = t >> 5;
  const int lane = t & 31;
  const int n0 = blockIdx.x * 16;           // 3125 blocks
  const int l16 = lane & 15, kh = lane >> 4;

  // staging role: row sn (0..15) of the weight tile, k-step sc (0..15)
  const int sn = t >> 4;                    // 2 rows per wave -> ~2 lines/instr
  const int sc = t & 15;
  const float* wrow = W + (size_t)(n0 + sn) * 1024 + sc * 4;

  const float* a0row = hid + (wave * 32 + l16) * 1024;
  const float* a1row = a0row + 16 * 1024;

  v8f acc0 = {}, acc1 = {};

  // prologue: stage chunk 0 into buffer 0
#pragma unroll
  for (int q = 0; q < 2; ++q)
    stage_b64(wrow + 2 * q, &sb[0][(sc * 32 + q * 16 + sn) * 2]);

  int buf = 0;
  for (int kc = 0; kc < 1024; kc += KC) {
    const bool more = (kc + KC) < 1024;
    if (more) {
#pragma unroll
      for (int q = 0; q < 2; ++q)
        stage_b64(wrow + kc + KC + 2 * q, &sb[buf ^ 1][(sc * 32 + q * 16 + sn) * 2]);
    }
#if USE_ASYNC
    if (more) __builtin_amdgcn_s_wait_asynccnt(2);
    else      __builtin_amdgcn_s_wait_asynccnt(0);
#endif
    __syncthreads();

    const float* bfrag = &sb[buf][lane * 2];
    const float* a0 = a0row + kc + 2 * kh;
    const float* a1 = a1row + kc + 2 * kh;
#pragma unroll 4
    for (int s = 0; s < 16; ++s) {
      v2f b  = *(const v2f*)(bfrag + s * 64);
      v2f va0 = *(const v2f*)(a0 + s * 4);
      v2f va1 = *(const v2f*)(a1 + s * 4);
      acc0 = wmma4(va0, b, acc0);
      acc1 = wmma4(va1, b, acc1);
    }
    __syncthreads();
    buf ^= 1;
  }

  const float bb = bias[n0 + l16];
  const int m0 = wave * 32;
#pragma unroll
  for (int r = 0; r < 8; ++r) {
    out[(size_t)(m0 + r + 8 * kh) * VP + n0 + l16] = acc0[r] + bb;
    out[(size_t)(m0 + 16 + r + 8 * kh) * VP + n0 + l16] = acc1[r] + bb;
  }
}

// ---------------------------------------------------------------------------
// Kernel 7: per-row max and sum(exp(x-max)) over 50000 logits (in d_out).
// ---------------------------------------------------------------------------
__global__ __launch_bounds__(256) void row_stats(const float* __restrict__ out,
                                                 float* __restrict__ rmax,
                                                 float* __restrict__ rsum) {
  __shared__ float red[256];
  const int b = blockIdx.x;
  const int t = threadIdx.x;
  const float* row = out + (size_t)b * VP;

  float m = -3.4e38f;
  for (int j = t; j < VN; j += 256) m = fmaxf(m, row[j]);
  red[t] = m; __syncthreads();
  for (int s = 128; s > 0; s >>= 1) {
    if (t < s) red[t] = fmaxf(red[t], red[t + s]);
    __syncthreads();
  }
  m = red[0];
  __syncthreads();

  float sum = 0.f;
  for (int j = t; j < VN; j += 256) sum += expf(row[j] - m);
  red[t] = sum; __syncthreads();
  for (int s = 128; s > 0; s >>= 1) {
    if (t < s) red[t] += red[t + s];
    __syncthreads();
  }
  if (t == 0) { rmax[b] = m; rsum[b] = red[0]; }
}

// ---------------------------------------------------------------------------
// Kernel 8: in-place p = gen * softmax(logits); zero the OOV tail columns.
// ---------------------------------------------------------------------------
__global__ __launch_bounds__(256) void finalize(float* __restrict__ out,
                                                const float* __restrict__ rmax,
                                                const float* __restrict__ rsum,
                                                const float* __restrict__ gen) {
  const int b = blockIdx.x;
  const int t = threadIdx.x;
  float* row = out + (size_t)b * VP;
  const float m = rmax[b];
  const float scale = gen[b] / rsum[b];
  for (int j = t; j < VN; j += 256) row[j] = expf(row[j] - m) * scale;
  if (t < OOVN) row[VN + t] = 0.0f;
}

// ---------------------------------------------------------------------------
// Kernel 9: scatter-add pointer-copy mass: p[b, ids[b,a]] += (1-gen[b])*att[b,a]
// ---------------------------------------------------------------------------
__global__ __launch_bounds__(256) void scatter_add(float* __restrict__ out,
                                                   const int* __restrict__ ids,
                                                   const float* __restrict__ att,
                                                   const float* __restrict__ gen) {
  const int idx = blockIdx.x * 256 + threadIdx.x;   // 256*400
  const int b = idx / AN;
  const int a = idx - b * AN;
  const float val = (1.0f - gen[b]) * att[idx];
  atomicAdd(&out[(size_t)b * VP + ids[idx]], val);
}

// ---------------------------------------------------------------------------
extern "C" void kernel_launch(void* const* d_in, const int* in_sizes, int n_in,
                              void* d_out, int out_size, void* d_ws,
                              size_t ws_size, hipStream_t stream) {
  const float* x      = (const float*)d_in[0];   // (256,1,256) -> x0
  const float* eo     = (const float*)d_in[2];   // (256,400,400)
  const float* estate = (const float*)d_in[4];   // (256,512)
  const int*   ids    = (const int*)d_in[6];     // (256,400)
  const float* Wih    = (const float*)d_in[7];   // (2048,256)
  const float* bih    = (const float*)d_in[9];
  const float* bhh    = (const float*)d_in[10];
  const float* Whw    = (const float*)d_in[11];  // (400,512)
  const float* Whb    = (const float*)d_in[12];
  const float* Wsw    = (const float*)d_in[13];  // (400,512)
  const float* Wsb    = (const float*)d_in[14];
  const float* vvec   = (const float*)d_in[15];  // (400,)
  const float* fc1w   = (const float*)d_in[16];  // (1024,912)
  const float* fc1b   = (const float*)d_in[17];
  const float* fc2w   = (const float*)d_in[18];  // (50000,1024)
  const float* fc2b   = (const float*)d_in[19];
  const float* pg1    = (const float*)d_in[20];  // (1,256)
  const float* pg2    = (const float*)d_in[21];  // (1,400)
  const float* pg3    = (const float*)d_in[22];  // (1,512)
  float* out = (float*)d_out;

  float* ws    = (float*)d_ws;
  float* gates = ws;                         // 256*2048
  float* h     = gates + BN * G4H;           // 256*512
  float* e     = h + BN * HN;                // 256*400
  float* att   = e + BN * AN;                // 256*400
  float* ctx   = att + BN * AN;              // 256*400
  float* hid   = ctx + BN * AN;              // 256*1024
  float* gen   = hid + BN * 1024;            // 256
  float* rmax  = gen + BN;                   // 256
  float* rsum  = rmax + BN;                  // 256

  gates_gemm<<<256, 256, 0, stream>>>(x, Wih, bih, bhh, gates);
  lstm_act<<<(BN * HN) / 256, 256, 0, stream>>>(gates, h);
  attn_e<<<50, 256, 0, stream>>>(estate, Whw, Whb, h, Wsw, Wsb, e);
  attn_ctx<<<BN, 256, 0, stream>>>(e, vvec, eo, x, h, pg1, pg2, pg3, att, ctx, gen);
  fc1_gemm<<<128, 256, 0, stream>>>(ctx, h, fc1w, fc1b, hid);
  fc2_gemm<<<VN / 16, 256, 0, stream>>>(hid, fc2w, fc2b, out);
  row_stats<<<BN, 256, 0, stream>>>(out, rmax, rsum);
  finalize<<<BN, 256, 0, stream>>>(out, rmax, rsum, gen);
  scatter_add<<<(BN * AN) / 256, 256, 0, stream>>>(out, ids, att, gen);
}